// FractalDim2d_44555990729224
// MI455X (gfx1250) — compile-verified
//
#include <hip/hip_runtime.h>

typedef __attribute__((ext_vector_type(2))) float v2f;
typedef __attribute__((ext_vector_type(8))) float v8f;

// ---- workspace layout (float offsets) ----
// pyramid levels 1..7, then accumulators
#define GP_OFF    349504            // 8 levels * 112 moment slots
#define GS1_OFF   (GP_OFF + 896)    // 8 floats: per-level sum(x)
#define GSXL_OFF  (GP_OFF + 904)    // 8 floats: per-level sum(x*log x)
#define ACC_FLOATS 912
#define ASTRIDE 20                  // LDS row stride (floats): 16B aligned, spreads banks

__global__ void zero_acc_kernel(float* ws) {
  int i = blockIdx.x * blockDim.x + threadIdx.x;
  if (i < ACC_FLOATS) ws[GP_OFF + i] = 0.0f;
}

// 2x2 sum pooling: out[Wo x Wo] from in[2Wo x 2Wo]
__global__ void pool2_kernel(const float* __restrict__ in, float* __restrict__ out, int Wo) {
  int i = blockIdx.x * blockDim.x + threadIdx.x;
  int n = Wo * Wo;
  if (i >= n) return;
  int y = i / Wo;
  int x = i - y * Wo;
  int Wi = Wo << 1;
  const float* r0 = in + (size_t)(2 * y) * Wi + 2 * x;
  const float* r1 = r0 + Wi;
  out[i] = (r0[0] + r0[1]) + (r1[0] + r1[1]);
}

// Per-level moment kernel. For each element x: t = x^0.09; accumulate
// P[16a+b] += t^(16a) * t^b for a=0..6, b=0..15 via V_WMMA_F32_16X16X4_F32
// (A[i,a]=u^a as 16x4 tile columns, B[i,b]=t^b rows; C = 16x16 f32 accum).
__global__ __launch_bounds__(256) void moments_kernel(
    const float* __restrict__ lvl, int numel,
    float* __restrict__ gP, float* __restrict__ gS1, float* __restrict__ gSXL) {
  __shared__ float Ash[8][32 * ASTRIDE];
  __shared__ float Bsh[8][32 * ASTRIDE];
  __shared__ float accP[112];
  __shared__ float accS[2];

  const int tid  = threadIdx.x;
  const int wave = tid >> 5;
  const int lane = tid & 31;
  const int half = lane >> 4;   // 0: lanes 0-15, 1: lanes 16-31
  const int lm   = lane & 15;

  if (tid < 112) accP[tid] = 0.0f;
  if (tid < 2)   accS[tid] = 0.0f;
  __syncthreads();

  float* Al = Ash[wave];
  float* Bl = Bsh[wave];

  v8f c = {0.f, 0.f, 0.f, 0.f, 0.f, 0.f, 0.f, 0.f};
  float s1 = 0.0f, sxl = 0.0f;

  const int totalWaves = gridDim.x * 8;
  const int ntiles = (numel + 31) >> 5;
  const int T = (ntiles + totalWaves - 1) / totalWaves;  // uniform trip count
  const int gwave = blockIdx.x * 8 + wave;

  for (int it = 0; it < T; ++it) {
    const int tile = it * totalWaves + gwave;
    const int idx  = tile * 32 + lane;
    float x = 0.0f;
    if (tile < ntiles && idx < numel) x = lvl[idx];
    const bool valid = (x != 0.0f);          // reference masks zeros
    float t = 0.0f;
    if (valid) {
      float lx = __logf(x);
      t = __expf(0.09f * lx);                // t = x^(9/100)
      s1  += x;
      sxl += x * lx;
    }
    // B row for this element: t^0..t^15 (0 if masked, incl. t^0)
    float pb[16];
    pb[0] = valid ? 1.0f : 0.0f;
    #pragma unroll
    for (int b = 1; b < 16; ++b) pb[b] = pb[b - 1] * t;
    // A column: u^0..u^6, u = t^16; rows 7..15 zero-padded
    const float u = pb[15] * t;
    float ua[7];
    ua[0] = pb[0];
    #pragma unroll
    for (int a = 1; a < 7; ++a) ua[a] = ua[a - 1] * u;

    float4* Bv = reinterpret_cast<float4*>(&Bl[lane * ASTRIDE]);
    Bv[0] = make_float4(pb[0],  pb[1],  pb[2],  pb[3]);
    Bv[1] = make_float4(pb[4],  pb[5],  pb[6],  pb[7]);
    Bv[2] = make_float4(pb[8],  pb[9],  pb[10], pb[11]);
    Bv[3] = make_float4(pb[12], pb[13], pb[14], pb[15]);
    float4* Av = reinterpret_cast<float4*>(&Al[lane * ASTRIDE]);
    Av[0] = make_float4(ua[0], ua[1], ua[2], ua[3]);
    Av[1] = make_float4(ua[4], ua[5], ua[6], 0.0f);
    Av[2] = make_float4(0.0f, 0.0f, 0.0f, 0.0f);
    Av[3] = make_float4(0.0f, 0.0f, 0.0f, 0.0f);
    __syncthreads();

    // 8 WMMAs consume the wave's 32 staged elements (K=4 each).
    // A 16x4 f32 layout: lane<16 -> K={0,1}, lane>=16 -> K={2,3}; M = lane%16.
    // B 4x16 mirrors it; element e = 4j + K lives at row e of the LDS tile.
    #pragma unroll
    for (int j = 0; j < 8; ++j) {
      const int i0 = (4 * j + 2 * half) * ASTRIDE + lm;
      v2f av, bv;
      av[0] = Al[i0];
      av[1] = Al[i0 + ASTRIDE];
      bv[0] = Bl[i0];
      bv[1] = Bl[i0 + ASTRIDE];
      c = __builtin_amdgcn_wmma_f32_16x16x4_f32(
          /*neg_a=*/false, av, /*neg_b=*/false, bv,
          /*c_mod=*/(short)0, c, /*reuse_a=*/false, /*reuse_b=*/false);
    }
    __syncthreads();
  }

  // wave32 reduction of scalar sums
  #pragma unroll
  for (int o = 16; o > 0; o >>= 1) {
    s1  += __shfl_xor(s1,  o, 32);
    sxl += __shfl_xor(sxl, o, 32);
  }
  if (lane == 0) { atomicAdd(&accS[0], s1); atomicAdd(&accS[1], sxl); }

  // C layout: VGPR r, lanes 0-15 -> M=r (a=r), N=lane. Rows a>=7 are zero.
  if (half == 0) {
    #pragma unroll
    for (int r = 0; r < 7; ++r) atomicAdd(&accP[r * 16 + lm], c[r]);
  }
  __syncthreads();
  if (tid < 112) atomicAdd(&gP[tid], accP[tid]);
  if (tid == 0)  atomicAdd(gS1,  accS[0]);
  if (tid == 1)  atomicAdd(gSXL, accS[1]);
}

__global__ void finalize_kernel(const float* __restrict__ ws, const float* __restrict__ bounds,
                                float* __restrict__ out) {
  const int k = threadIdx.x;
  if (k >= 100) return;
  const float LN2f = 0.69314718055994530942f;
  const float* gP   = ws + GP_OFF;
  const float* gS1  = ws + GS1_OFF;
  const float* gSXL = ws + GSXL_OFF;
  const float b  = gS1[0];                 // sum of nonzero level-0 pixels
  const float lb = __logf(b);
  const float bd = bounds[k];
  const bool is1 = (bd == 1.0f);
  const float a  = is1 ? 1.0f : 1.0f / (bd - 1.0f);
  float a01 = 0.f, pq = 0.f, pp = 0.f;
  #pragma unroll
  for (int m = 0; m < 8; ++m) {
    // entropy branch: sum (x/b) log(x/b) = (sum x log x - log b * sum x)/b
    float P = is1 ? ((gSXL[m] - lb * gS1[m]) / b) : gP[m * 112 + k];
    a01 += P;
    pq  += P * ((float)m * LN2f);
    pp  += P * P;
  }
  const float qs  = 28.0f * LN2f;          // sum_{m=0..7} m*ln2
  const float num = 8.0f * pq - a01 * qs;  // a00 = n = 8
  const float den = 8.0f * pp - a01 * a01;
  out[k] = a * num / den;
}

extern "C" void kernel_launch(void* const* d_in, const int* in_sizes, int n_in,
                              void* d_out, int out_size, void* d_ws, size_t ws_size,
                              hipStream_t stream) {
  const float* img    = (const float*)d_in[0];
  const float* bounds = (const float*)d_in[1];
  float* out = (float*)d_out;
  float* ws  = (float*)d_ws;
  (void)in_sizes; (void)n_in; (void)out_size; (void)ws_size;

  zero_acc_kernel<<<1, 1024, 0, stream>>>(ws);

  // pyramid: level i at pyrOff[i] (floats), level 0 is the input
  static const size_t pyrOff[8] = {0, 0, 262144, 327680, 344064, 348160, 349184, 349440};
  const float* levels[8];
  levels[0] = img;
  {
    const float* src = img;
    int W = 1024;
    for (int i = 1; i < 8; ++i) {
      const int Wo = W >> 1;
      float* dst = ws + pyrOff[i];
      const int n = Wo * Wo;
      pool2_kernel<<<(n + 255) / 256, 256, 0, stream>>>(src, dst, Wo);
      levels[i] = dst;
      src = dst;
      W = Wo;
    }
  }

  for (int lev = 0; lev < 8; ++lev) {
    const int W = 1024 >> lev;
    const int numel = W * W;
    const int ntiles = (numel + 31) / 32;
    int blocks = (ntiles + 7) / 8;
    if (blocks > 512) blocks = 512;
    moments_kernel<<<blocks, 256, 0, stream>>>(
        levels[lev], numel,
        ws + GP_OFF + lev * 112, ws + GS1_OFF + lev, ws + GSXL_OFF + lev);
  }

  finalize_kernel<<<1, 128, 0, stream>>>(ws, bounds, out);
}